// UniCascadeBboxNet_59897613910776
// MI455X (gfx1250) — compile-verified
//
#include <hip/hip_runtime.h>

// CDNA5 / gfx1250: wave32, WMMA 16x16x4 F32 used as a cross-lane 4-tap
// weighted-reduction engine for bilinear interpolation.
//
// Perf model (MI455X): feat_map = 128 MiB fits in the 192 MB global L2, so
// after first touch every bilinear gather is an L2 hit. HBM traffic ~= 128 MiB
// (feat first touch) + 72 MiB (output, written once). Output stores are
// non-temporal so the write stream does not evict the resident feature map.

typedef float v2f __attribute__((ext_vector_type(2)));
typedef float v8f __attribute__((ext_vector_type(8)));

#define B_   8
#define C_   256
#define H_   128
#define W_   128
#define P_   9
#define HW_  (H_ * W_)        // 16384
#define CHW_ (C_ * HW_)       // 4194304
#define WAVES_PER_BLOCK 8

__global__ __launch_bounds__(256)
void deform_bilinear_wmma(const float* __restrict__ feat,
                          const float* __restrict__ rois,
                          const float* __restrict__ offs,
                          const int*   __restrict__ stride_p,
                          float* __restrict__ out,
                          int S)
{
    const int lane = threadIdx.x & 31;
    // Force the sample id into an SGPR: all per-sample math scalarizes,
    // roi/offset reads become s_load, and the trip guard is a scalar branch
    // (EXEC stays all-1s -> legal around WMMA).
    const int s = __builtin_amdgcn_readfirstlane(
        (int)blockIdx.x * WAVES_PER_BLOCK + ((int)threadIdx.x >> 5));
    if (s >= S) return;

    const int n = s / P_;
    const int p = s - n * P_;

    const float inv_stride = 1.0f / (float)stride_p[0];

    // ---- per-sample scalar math (matches reference exactly) ----
    const float* roi = rois + (size_t)n * 5;
    const int   b  = (int)roi[0];
    const float x1 = roi[1], y1 = roi[2], x2 = roi[3], y2 = roi[4];
    const float cx = (x1 + x2) * 0.5f;
    const float cy = (y1 + y2) * 0.5f;
    const float rw = x2 - x1 + 1.0f;
    const float rh = y2 - y1 + 1.0f;

    const float* op = offs + (size_t)n * (2 * P_) + 2 * p;
    const float wcoord = (cx + op[0] * rw * 0.1f) * inv_stride;  // index[...,0] = w
    const float hcoord = (cy + op[1] * rh * 0.1f) * inv_stride;  // index[...,1] = h

    // bilinear corners + weights (reference semantics, incl. edge clamp rules)
    const float hlf   = fminf(fmaxf(floorf(hcoord), 0.0f), (float)(H_ - 1));
    const float hused = (hlf >= (float)(H_ - 1)) ? hlf : hcoord;
    const float hhf   = (hlf >= (float)(H_ - 1)) ? hlf : hlf + 1.0f;
    const float wlf   = fminf(fmaxf(floorf(wcoord), 0.0f), (float)(W_ - 1));
    const float wused = (wlf >= (float)(W_ - 1)) ? wlf : wcoord;
    const float whf   = (wlf >= (float)(W_ - 1)) ? wlf : wlf + 1.0f;

    const int hl = (int)hlf, hh = (int)hhf, wl = (int)wlf, wh = (int)whf;
    const float lh = hused - hlf, lw = wused - wlf;
    const float uh = 1.0f - lh,   uw = 1.0f - lw;
    const float w1 = uh * uw;   // (hl, wl)
    const float w2 = uh * lw;   // (hl, wh)
    const float w3 = lh * uw;   // (hh, wl)
    const float w4 = lh * lw;   // (hh, wh)

    // ---- A matrix: 16x4 f32, broadcast weights to all 16 rows ----
    // ISA layout: lanes 0-15 hold K=0 (vgpr0), K=1 (vgpr1);
    //             lanes 16-31 hold K=2 (vgpr0), K=3 (vgpr1).
    const int sel  = lane >> 4;    // 0: corners (v1,v2) on row hl ; 1: (v3,v4) on row hh
    const int cidx = lane & 15;    // channel within 16-wide chunk
    v2f a;
    a.x = sel ? w3 : w1;
    a.y = sel ? w4 : w2;

    // Per-lane gather base: channel cidx, row (hl|hh), col wl. p1 = right neighbor.
    const int dw = wh - wl;        // 0 or 1
    const float* p0 = feat + (size_t)b * CHW_
                           + (size_t)(sel ? hh : hl) * W_ + wl
                           + (size_t)cidx * HW_;
    const float* p1 = p0 + dw;

    float* outp = out + (size_t)s * C_ + lane;

    // 256 channels, 32 per iteration: lanes 0-15 take the "lo" 16 channels
    // (row M=0 of D_lo), lanes 16-31 take the "hi" 16 channels (row M=8 of
    // D_hi) -> one fully coalesced non-temporal b32 store per iteration.
#pragma unroll
    for (int c0 = 0; c0 < C_; c0 += 32) {
        v2f blo, bhi;
        blo.x = p0[(size_t)c0 * HW_];          // K = 2*sel + 0 corner value
        blo.y = p1[(size_t)c0 * HW_];          // K = 2*sel + 1 corner value
        bhi.x = p0[(size_t)(c0 + 16) * HW_];
        bhi.y = p1[(size_t)(c0 + 16) * HW_];

        v8f acc = {};
        v8f dlo = __builtin_amdgcn_wmma_f32_16x16x4_f32(
            false, a, false, blo, (short)0, acc, false, false);
        v8f dhi = __builtin_amdgcn_wmma_f32_16x16x4_f32(
            false, a, false, bhi, (short)0, acc, false, false);

        const float val = sel ? dhi[0] : dlo[0];
        __builtin_nontemporal_store(val, outp + c0);   // TH=NT: don't evict feat from L2
    }
}

extern "C" void kernel_launch(void* const* d_in, const int* in_sizes, int n_in,
                              void* d_out, int out_size, void* d_ws, size_t ws_size,
                              hipStream_t stream)
{
    const float* feat   = (const float*)d_in[0];
    const float* rois   = (const float*)d_in[1];
    const float* offs   = (const float*)d_in[2];
    const int*   stride = (const int*)d_in[3];
    // d_in[4] = num_point (fixed at 9 by the problem geometry)
    float* out = (float*)d_out;

    const int N = in_sizes[1] / 5;       // rois: [N,5]
    const int S = N * P_;
    const int blocks = (S + WAVES_PER_BLOCK - 1) / WAVES_PER_BLOCK;

    deform_bilinear_wmma<<<blocks, 256, 0, stream>>>(feat, rois, offs, stride, out, S);
}